// ChannelKiller_54365696033605
// MI455X (gfx1250) — compile-verified
//
#include <hip/hip_runtime.h>
#include <cstdint>

// 128-bit vector types
typedef float __attribute__((ext_vector_type(4))) v4f;
typedef int   __attribute__((vector_size(16)))    v4i;   // matches builtin param type

#define AS1 __attribute__((address_space(1)))
#define AS3 __attribute__((address_space(3)))

#if defined(__gfx1250__) &&                                                  \
    __has_builtin(__builtin_amdgcn_global_load_async_to_lds_b128) &&         \
    __has_builtin(__builtin_amdgcn_global_store_async_from_lds_b128) &&      \
    __has_builtin(__builtin_amdgcn_s_wait_asynccnt)
#define USE_ASYNC_LDS 1
#else
#define USE_ASYNC_LDS 0
#endif

// Tensor geometry: x is [16, 8, 1048576] fp32.
// In float4 units: channel = 2^18, batch = 8*2^18 = 2^21.
static constexpr unsigned kC4      = 1u << 18;          // float4 per channel
static constexpr unsigned kB4      = 1u << 21;          // float4 per batch
static constexpr unsigned kZeroPB  = 7u << 18;          // float4 to zero per batch (ch 1..7)
static constexpr unsigned kZeroTot = 16u * (7u << 18);  // 29,360,128 float4
static constexpr unsigned kCopyTot = 16u << 18;         // 4,194,304 float4 (channel 0)

// ---------------------------------------------------------------------------
// Kernel 1: zero channels 1..7 (448 MiB of NT B128 stores, no reads).
// Per batch b the zero span is contiguous: [b*kB4 + kC4, b*kB4 + kB4).
// ---------------------------------------------------------------------------
__global__ void __launch_bounds__(256)
zero_fill_kernel(v4f* __restrict__ out4) {
  const v4f z = {0.f, 0.f, 0.f, 0.f};
  const unsigned stride = gridDim.x * blockDim.x;
  for (unsigned i = blockIdx.x * blockDim.x + threadIdx.x; i < kZeroTot; i += stride) {
    const unsigned b = i / kZeroPB;          // const divide -> mulhi
    const unsigned r = i - b * kZeroPB;
    __builtin_nontemporal_store(z, out4 + (b * kB4 + kC4 + r));  // global_store_b128 th:NT
  }
}

// ---------------------------------------------------------------------------
// Kernel 2: copy channel 0 (64 MiB in, 64 MiB out).
// gfx1250 path: async global->LDS->global B128 copies tracked by ASYNCcnt,
// double-buffered so the next tile's load overlaps the current tile's store.
// Each thread owns its private LDS slot, so only s_wait_asynccnt is needed
// (no workgroup barriers).
// ---------------------------------------------------------------------------
__global__ void __launch_bounds__(256)
ch0_copy_kernel(const v4f* __restrict__ in4, v4f* __restrict__ out4) {
  const unsigned tid    = threadIdx.x;
  const unsigned stride = gridDim.x * blockDim.x;
  unsigned j = blockIdx.x * blockDim.x + tid;   // index into channel-0 float4 space

#if USE_ASYNC_LDS
  __shared__ v4f buf[2][256];
  unsigned stage = 0;
  if (j < kCopyTot) {
    const unsigned flat = ((j >> 18) << 21) | (j & (kC4 - 1u));
    __builtin_amdgcn_global_load_async_to_lds_b128(
        (AS1 v4i*)(in4 + flat), (AS3 v4i*)&buf[0][tid], 0, 0);
  }
  while (j < kCopyTot) {
    __builtin_amdgcn_s_wait_asynccnt(0);        // load(j) (and store(j-stride)) done
    const unsigned flat = ((j >> 18) << 21) | (j & (kC4 - 1u));
    __builtin_amdgcn_global_store_async_from_lds_b128(
        (AS1 v4i*)(out4 + flat), (AS3 v4i*)&buf[stage][tid], 0, 0);
    const unsigned jn = j + stride;
    if (jn < kCopyTot) {                         // prefetch next tile into other buffer,
      const unsigned flatn = ((jn >> 18) << 21) | (jn & (kC4 - 1u));
      __builtin_amdgcn_global_load_async_to_lds_b128(  // overlaps the store above
          (AS1 v4i*)(in4 + flatn), (AS3 v4i*)&buf[stage ^ 1][tid], 0, 0);
    }
    stage ^= 1;
    j = jn;
  }
  __builtin_amdgcn_s_wait_asynccnt(0);
#else
  for (; j < kCopyTot; j += stride) {
    const unsigned flat = ((j >> 18) << 21) | (j & (kC4 - 1u));
    v4f v = __builtin_nontemporal_load(in4 + flat);
    __builtin_nontemporal_store(v, out4 + flat);
  }
#endif
}

// ---------------------------------------------------------------------------
// Host launcher
// ---------------------------------------------------------------------------
extern "C" void kernel_launch(void* const* d_in, const int* in_sizes, int n_in,
                              void* d_out, int out_size, void* d_ws, size_t ws_size,
                              hipStream_t stream) {
  (void)in_sizes; (void)n_in; (void)out_size; (void)d_ws; (void)ws_size;
  const v4f* x4   = (const v4f*)d_in[0];
  v4f*       out4 = (v4f*)d_out;

  // Zero-fill: 29,360,128 float4s; 8 stores/thread -> 14336 blocks of 256.
  zero_fill_kernel<<<14336, 256, 0, stream>>>(out4);

  // Channel-0 copy: 4,194,304 float4s; 8 tiles/thread -> 2048 blocks of 256.
  ch0_copy_kernel<<<2048, 256, 0, stream>>>(x4, out4);
}